// AttentionHead_39195871543648
// MI455X (gfx1250) — compile-verified
//
#include <hip/hip_runtime.h>
#include <stdint.h>

// CDNA5 / gfx1250: wave32, WMMA f32 16x16x4 (full-precision fp32 matrix path),
// async global->LDS DMA (ASYNCcnt) for double-buffered tile staging.
typedef float v2f __attribute__((ext_vector_type(2)));
typedef float v8f __attribute__((ext_vector_type(8)));

#define HIDDEN 768
#define HEAD_DIM 64
#define BATCH 8
#define SEQ 2048

__device__ __forceinline__ v8f wmma_f32_k4(v2f a, v2f b, v8f c) {
  // 8 args: (neg_a, A, neg_b, B, c_mod, C, reuse_a, reuse_b)
  return __builtin_amdgcn_wmma_f32_16x16x4_f32(false, a, false, b, (short)0, c,
                                               false, false);
}

// Async DMA: copies 16B per enabled lane from global to LDS, tracked by
// ASYNCcnt (no VGPR data round-trip).  lds_off = wave-relative LDS byte addr
// (low 32 bits of the generic pointer to a __shared__ object).
__device__ __forceinline__ void async_copy_b128(uint32_t lds_off,
                                                const float* gptr) {
  asm volatile("global_load_async_to_lds_b128 %0, %1, off"
               :
               : "v"(lds_off), "v"(gptr)
               : "memory");
}
__device__ __forceinline__ void wait_async0() {
  asm volatile("s_wait_asynccnt 0x0" ::: "memory");
}
__device__ __forceinline__ uint32_t lds_addr(const void* p) {
  return (uint32_t)(uintptr_t)p;
}

// ---------------------------------------------------------------------------
// Kernel 1: P = X[M,768] * W[768,64] + bias.  M = B*S = 16384.
// Block = 256 threads (8 waves); 32 rows x 64 cols of P per block.
// Double-buffered async global->LDS staging of X/W K-chunks (64 wide).
// ---------------------------------------------------------------------------
__global__ __launch_bounds__(256) void proj_kernel(
    const float* __restrict__ X, const float* __restrict__ W,
    const float* __restrict__ bias, float* __restrict__ P) {
  __shared__ float Xs[2][32][68];  // pad 4: bank step 4/row, conflict-free
  __shared__ float Ws[2][64][68];

  const int tid  = threadIdx.x;
  const int wave = tid >> 5;
  const int lane = tid & 31;
  const int half = lane >> 4;
  const int l    = lane & 15;
  const int rt   = wave >> 2;
  const int ct   = wave & 3;
  const int row0 = blockIdx.x * 32;

  auto issue = [&](int buf, int kc) {
#pragma unroll
    for (int i = 0; i < 2; ++i) {  // X tile: 32x64 = 512 float4
      int idx = tid + i * 256;
      int r = idx >> 4, c4 = idx & 15;
      async_copy_b128(lds_addr(&Xs[buf][r][c4 * 4]),
                      &X[(size_t)(row0 + r) * HIDDEN + kc * 64 + c4 * 4]);
    }
#pragma unroll
    for (int i = 0; i < 4; ++i) {  // W tile: 64x64 = 1024 float4
      int idx = tid + i * 256;
      int r = idx >> 4, c4 = idx & 15;
      async_copy_b128(lds_addr(&Ws[buf][r][c4 * 4]),
                      &W[(size_t)(kc * 64 + r) * HEAD_DIM + c4 * 4]);
    }
  };

  v8f acc = {};
  issue(0, 0);

#pragma unroll 1
  for (int kc = 0; kc < HIDDEN / 64; ++kc) {
    const int cur = kc & 1;
    wait_async0();     // my wave's DMA for buf[cur] done
    __syncthreads();   // everyone's done; prior readers of buf[cur^1] drained
    if (kc + 1 < HIDDEN / 64) issue(cur ^ 1, kc + 1);

#pragma unroll
    for (int kk = 0; kk < 16; ++kk) {
      const int k = kk * 4 + half * 2;
      v2f a, b;
      a.x = Xs[cur][rt * 16 + l][k];
      a.y = Xs[cur][rt * 16 + l][k + 1];
      b.x = Ws[cur][k][ct * 16 + l];
      b.y = Ws[cur][k + 1][ct * 16 + l];
      acc = wmma_f32_k4(a, b, acc);
    }
    __syncthreads();   // done reading buf[cur] before it is re-filled
  }

  // D layout: VGPR j -> row j (lanes 0-15) / row j+8 (lanes 16-31), col = l
  const int col = ct * 16 + l;
  const float bv = bias[col];
#pragma unroll
  for (int j = 0; j < 8; ++j) {
    const int row = row0 + rt * 16 + j + half * 8;
    P[(size_t)row * HEAD_DIM + col] = acc[j] + bv;
  }
}

// ---------------------------------------------------------------------------
// Kernel 2: flash attention. 4 waves/block, one 16-query tile per wave.
// K/V 16x64 tiles staged once per block via async DMA (double-buffered) and
// shared by all 4 waves.  Per key tile: 16 WMMA (Q.K^T) + online softmax +
// 16 WMMA (P.V).
// ---------------------------------------------------------------------------
#define APB 4  // waves (query tiles) per block

__global__ __launch_bounds__(32 * APB) void attn_kernel(
    const float* __restrict__ Q, const float* __restrict__ K,
    const float* __restrict__ V, const int* __restrict__ mask,
    float* __restrict__ O) {
  __shared__ float Kt[2][16][68];
  __shared__ float Vt[2][16][68];
  __shared__ float Pbuf[APB][16][17];  // per-wave D->A transpose buffer

  const int tid  = threadIdx.x;
  const int wave = tid >> 5;
  const int lane = tid & 31;
  const int half = lane >> 4;
  const int l    = lane & 15;
  const int b    = blockIdx.y;
  const int qrow0 = (blockIdx.x * APB + wave) * 16;

  const float* __restrict__ Qb = Q + ((size_t)b * SEQ + qrow0) * HEAD_DIM;
  const float* __restrict__ Kb = K + (size_t)b * SEQ * HEAD_DIM;
  const float* __restrict__ Vb = V + (size_t)b * SEQ * HEAD_DIM;
  const int* __restrict__ mb = mask + (size_t)b * SEQ;

  auto issueKV = [&](int buf, int key0) {
#pragma unroll
    for (int i = 0; i < 2; ++i) {  // 16x64 = 256 float4 each for K and V
      int idx = tid + i * 128;
      int r = idx >> 4, c4 = idx & 15;
      async_copy_b128(lds_addr(&Kt[buf][r][c4 * 4]),
                      &Kb[(size_t)(key0 + r) * HEAD_DIM + c4 * 4]);
      async_copy_b128(lds_addr(&Vt[buf][r][c4 * 4]),
                      &Vb[(size_t)(key0 + r) * HEAD_DIM + c4 * 4]);
    }
  };

  // Q tile (16x64) in A-operand layout: lane l holds row l; step kk holds
  // d = {4kk+2h, 4kk+2h+1}.  Loaded once, L2-resident.
  v2f qa[16];
#pragma unroll
  for (int kk = 0; kk < 16; ++kk) {
    const int d = kk * 4 + half * 2;
    qa[kk].x = Qb[(size_t)l * HEAD_DIM + d];
    qa[kk].y = Qb[(size_t)l * HEAD_DIM + d + 1];
  }

  const float scale = 0.125f;  // 1/sqrt(64)
  float mrow[8], lrow[8];
  v8f o[4];
#pragma unroll
  for (int j = 0; j < 8; ++j) { mrow[j] = -3.0e38f; lrow[j] = 0.0f; }
#pragma unroll
  for (int t = 0; t < 4; ++t) o[t] = (v8f){};

  issueKV(0, 0);

#pragma unroll 1
  for (int key0 = 0; key0 < SEQ; key0 += 16) {
    const int cur = (key0 >> 4) & 1;
    wait_async0();
    __syncthreads();
    if (key0 + 16 < SEQ) issueKV(cur ^ 1, key0 + 16);

    // --- S = Q.K^T (16 queries x 16 keys), reduce over d = 64 ---
    v8f s = {};
#pragma unroll
    for (int kk = 0; kk < 16; ++kk) {
      const int d = kk * 4 + half * 2;
      v2f kb;  // B layout: lane = key col, rows d / d+1
      kb.x = Kt[cur][l][d];
      kb.y = Kt[cur][l][d + 1];
      s = wmma_f32_k4(qa[kk], kb, s);
    }

    // --- online softmax; lane's key col = key0 + l, row = j + 8*half ---
    const float madd = (mb[key0 + l] == 0) ? -1.0e9f : 0.0f;
#pragma unroll
    for (int j = 0; j < 8; ++j) {
      float sj = s[j] * scale + madd;
      float mx = sj;  // row max across the 16 lanes of this half
#pragma unroll
      for (int off = 8; off >= 1; off >>= 1)
        mx = fmaxf(mx, __shfl_xor(mx, off, 32));
      const float mnew = fmaxf(mrow[j], mx);
      const float corr = __expf(mrow[j] - mnew);
      const float p    = __expf(sj - mnew);
      float ps = p;  // row sum
#pragma unroll
      for (int off = 8; off >= 1; off >>= 1) ps += __shfl_xor(ps, off, 32);
      lrow[j] = lrow[j] * corr + ps;
      mrow[j] = mnew;
#pragma unroll
      for (int t = 0; t < 4; ++t) o[t][j] *= corr;  // rescale accumulator
      Pbuf[wave][j + 8 * half][l] = p;  // stage for D->A transpose
    }

    // --- read P back in A-operand layout (intra-wave LDS RAW is in-order) ---
    v2f pa[4];
#pragma unroll
    for (int kk = 0; kk < 4; ++kk) {
      const int kc = kk * 4 + half * 2;
      pa[kk].x = Pbuf[wave][l][kc];
      pa[kk].y = Pbuf[wave][l][kc + 1];
    }

    // --- O += P.V (16 queries x 64 dims), reduce over 16 keys ---
#pragma unroll
    for (int t = 0; t < 4; ++t) {
#pragma unroll
      for (int kk = 0; kk < 4; ++kk) {
        const int kr = kk * 4 + half * 2;
        v2f vb;  // B layout: lane = v-col, rows kr / kr+1
        vb.x = Vt[cur][kr][t * 16 + l];
        vb.y = Vt[cur][kr + 1][t * 16 + l];
        o[t] = wmma_f32_k4(pa[kk], vb, o[t]);
      }
    }
    __syncthreads();  // done reading Kt/Vt[cur] before re-fill
  }

  // --- epilogue: O = o / l ---
#pragma unroll
  for (int t = 0; t < 4; ++t) {
#pragma unroll
    for (int j = 0; j < 8; ++j) {
      const int row = qrow0 + j + 8 * half;
      O[((size_t)b * SEQ + row) * HEAD_DIM + t * 16 + l] = o[t][j] / lrow[j];
    }
  }
}

// ---------------------------------------------------------------------------
extern "C" void kernel_launch(void* const* d_in, const int* in_sizes, int n_in,
                              void* d_out, int out_size, void* d_ws,
                              size_t ws_size, hipStream_t stream) {
  const float* query = (const float*)d_in[0];
  const float* key   = (const float*)d_in[1];
  const float* value = (const float*)d_in[2];
  const int*   mask  = (const int*)d_in[3];
  const float* Wq = (const float*)d_in[4];
  const float* bq = (const float*)d_in[5];
  const float* Wk = (const float*)d_in[6];
  const float* bk = (const float*)d_in[7];
  const float* Wv = (const float*)d_in[8];
  const float* bv = (const float*)d_in[9];
  float* out = (float*)d_out;
  (void)in_sizes; (void)n_in; (void)out_size; (void)ws_size;

  const size_t M = (size_t)BATCH * SEQ;  // 16384
  float* qp = (float*)d_ws;              // 3 x 4 MB scratch
  float* kp = qp + M * HEAD_DIM;
  float* vp = kp + M * HEAD_DIM;

  dim3 gp((unsigned)(M / 32)), bp(256);
  proj_kernel<<<gp, bp, 0, stream>>>(query, Wq, bq, qp);
  proj_kernel<<<gp, bp, 0, stream>>>(key,   Wk, bk, kp);
  proj_kernel<<<gp, bp, 0, stream>>>(value, Wv, bv, vp);

  dim3 ga(SEQ / 16 / APB, BATCH), ba(32 * APB);
  attn_kernel<<<ga, ba, 0, stream>>>(qp, kp, vp, mask, out);
}